// TemporalWaveletTransform_50912542326953
// MI455X (gfx1250) — compile-verified
//
#include <hip/hip_runtime.h>

typedef float v2f __attribute__((ext_vector_type(2)));
typedef float v8f __attribute__((ext_vector_type(8)));

#define FEAT 256
#define NB 16

// db4 decomposition filters, REVERSED (cross-correlation form used by the
// reference: out[o] = sum_k DEC[7-k] * ext[2o+k], ext[i] = x[sym(i-6)]).
#define RLO0 (0.23037781330885523f)
#define RLO1 (0.7148465705525415f)
#define RLO2 (0.6308807679295904f)
#define RLO3 (-0.02798376941698385f)
#define RLO4 (-0.18703481171888114f)
#define RLO5 (0.030841381835986965f)
#define RLO6 (0.032883011666982945f)
#define RLO7 (-0.010597401784997278f)

#define RHI0 (-0.010597401784997278f)
#define RHI1 (-0.032883011666982945f)
#define RHI2 (0.030841381835986965f)
#define RHI3 (0.18703481171888114f)
#define RHI4 (-0.02798376941698385f)
#define RHI5 (-0.6308807679295904f)
#define RHI6 (0.7148465705525415f)
#define RHI7 (-0.23037781330885523f)

// pywt 'symmetric' (half-sample) reflection; single reflection is enough
// since |overhang| <= 8 << L for every level here.
__device__ __forceinline__ int symref(int t, int L) {
  t = (t < 0) ? (-1 - t) : t;
  t = (t >= L) ? (2 * L - 1 - t) : t;
  return t;
}

__device__ __forceinline__ float pick4(int i, float a, float b, float c, float d) {
  float r = (i == 0) ? a : b;
  r = (i == 2) ? c : r;
  r = (i == 3) ? d : r;
  return r;
}

// One DWT level as banded GEMM on V_WMMA_F32_16X16X4_F32.
// Input layout: [B, Lin, FEAT] (f contiguous). Each wave produces a
// 16(o) x 16(f) tile for BOTH filters (lo -> outLo, hi -> outHi).
__global__ void __launch_bounds__(256)
dwt_level_wmma(const float* __restrict__ in, int Lin, int Lo, int nOT,
               float* __restrict__ outLo, unsigned long long pBLo, int pTLo,
               float* __restrict__ outHi, unsigned long long pBHi, int pTHi,
               int totalWaves)
{
  int wid = (int)((blockIdx.x * blockDim.x + threadIdx.x) >> 5);
  if (wid >= totalWaves) return;                 // wave-uniform; EXEC stays all-1s
  int lane = threadIdx.x & 31;
  int m  = lane & 15;                            // A row (M) == B/D column (N)
  int kh = lane >> 4;                            // lane-half selects K pair / D row half

  int ft = wid & 15;
  int ot = (wid >> 4) % nOT;
  int b  = wid / (16 * nOT);
  int f0 = ft * 16;
  int o_base = ot * 16;
  int t_start = 2 * o_base - 6;                  // first tap index (pre-reflection)

  // A fragment schedule: vgpr j covers band column trel = 4s + 2kh + j;
  // weight idx = trel - 2m is in [0,8) for exactly two steps: sA_j, sA_j+1.
  int c0  = 2 * m - 2 * kh;                      // j = 0
  int c1  = c0 - 1;                              // j = 1
  int sA0 = (c0 + 3) >> 2;
  int sA1 = (c1 + 3) >> 2;
  int i0  = 4 * sA0 - c0;                        // idx at step sA0, in [0,3]
  int i1  = 4 * sA1 - c1;

  float wl0_0 = pick4(i0, RLO0, RLO1, RLO2, RLO3);
  float wl1_0 = pick4(i0, RLO4, RLO5, RLO6, RLO7);
  float wl0_1 = pick4(i1, RLO0, RLO1, RLO2, RLO3);
  float wl1_1 = pick4(i1, RLO4, RLO5, RLO6, RLO7);
  float wh0_0 = pick4(i0, RHI0, RHI1, RHI2, RHI3);
  float wh1_0 = pick4(i0, RHI4, RHI5, RHI6, RHI7);
  float wh0_1 = pick4(i1, RHI0, RHI1, RHI2, RHI3);
  float wh1_1 = pick4(i1, RHI4, RHI5, RHI6, RHI7);

  const float* inb = in + (size_t)b * (size_t)Lin * FEAT + f0 + m;

  v8f accLo = {0.f, 0.f, 0.f, 0.f, 0.f, 0.f, 0.f, 0.f};
  v8f accHi = {0.f, 0.f, 0.f, 0.f, 0.f, 0.f, 0.f, 0.f};

#pragma unroll
  for (int s = 0; s < 10; ++s) {
    // B fragment 4x16: vgpr j holds row k = 2*kh + j, column n = m.
    int t0 = t_start + 4 * s + 2 * kh;
    int r0 = symref(t0, Lin);
    int r1 = symref(t0 + 1, Lin);
    v2f bf;
    bf.x = inb[(size_t)r0 * FEAT];
    bf.y = inb[(size_t)r1 * FEAT];

    v2f alo, ahi;
    alo.x = (s == sA0) ? wl0_0 : ((s == sA0 + 1) ? wl1_0 : 0.f);
    alo.y = (s == sA1) ? wl0_1 : ((s == sA1 + 1) ? wl1_1 : 0.f);
    ahi.x = (s == sA0) ? wh0_0 : ((s == sA0 + 1) ? wh1_0 : 0.f);
    ahi.y = (s == sA1) ? wh0_1 : ((s == sA1 + 1) ? wh1_1 : 0.f);

    accLo = __builtin_amdgcn_wmma_f32_16x16x4_f32(false, alo, false, bf,
                                                  (short)0, accLo, false, false);
    accHi = __builtin_amdgcn_wmma_f32_16x16x4_f32(false, ahi, false, bf,
                                                  (short)0, accHi, false, false);
  }

  // D layout: vgpr v -> row m = v + 8*kh, column n = lane&15.
  float* pLo = outLo + (size_t)b * pBLo + f0 + m;
  float* pHi = outHi + (size_t)b * pBHi + f0 + m;
#pragma unroll
  for (int v = 0; v < 8; ++v) {
    int o = o_base + v + 8 * kh;
    if (o < Lo) {
      pLo[(size_t)o * (size_t)pTLo] = accLo[v];
      pHi[(size_t)o * (size_t)pTHi] = accHi[v];
    }
  }
}

__global__ void __launch_bounds__(256)
zero_f4(float4* __restrict__ out, size_t n4)
{
  size_t i = (size_t)blockIdx.x * blockDim.x + threadIdx.x;
  if (i < n4) out[i] = make_float4(0.f, 0.f, 0.f, 0.f);
}

extern "C" void kernel_launch(void* const* d_in, const int* in_sizes, int n_in,
                              void* d_out, int out_size, void* d_ws, size_t ws_size,
                              hipStream_t stream) {
  (void)in_sizes; (void)n_in; (void)ws_size;
  const float* traj = (const float*)d_in[0];
  float* out = (float*)d_out;
  float* ws  = (float*)d_ws;

  const int T  = 4096;
  const int L1 = 2051, L2 = 1029, L3 = 518, L4 = 262;   // Lo per level

  // 1) Zero the whole [B,T,5,F] output (bands are zero-padded to T).
  size_t n4 = (size_t)out_size / 4;
  zero_f4<<<(unsigned)((n4 + 255) / 256), 256, 0, stream>>>((float4*)d_out, n4);

  // Output band addressing: elem(b,t,band,f) = ((b*T+t)*5 + band)*FEAT + f
  const unsigned long long PB = (unsigned long long)T * 5 * FEAT;  // per-batch pitch
  const int PT = 5 * FEAT;                                         // per-t pitch

  // cA intermediates in workspace, layout [B, Lo, FEAT] (~57 MB total).
  float* cA1 = ws;
  float* cA2 = cA1 + (size_t)NB * L1 * FEAT;
  float* cA3 = cA2 + (size_t)NB * L2 * FEAT;

  auto launch = [&](const float* inp, int Lin, int Lo,
                    float* lo, unsigned long long pBLo, int pTLo,
                    float* hi, unsigned long long pBHi, int pTHi) {
    int nOT = (Lo + 15) / 16;
    int totalWaves = NB * 16 * nOT;
    int blocks = (totalWaves + 7) / 8;
    dwt_level_wmma<<<blocks, 256, 0, stream>>>(inp, Lin, Lo, nOT,
                                               lo, pBLo, pTLo,
                                               hi, pBHi, pTHi, totalWaves);
  };

  // Band order [cA4, cD4, cD3, cD2, cD1] -> band index 0..4.
  launch(traj, T,  L1, cA1, (unsigned long long)L1 * FEAT, FEAT, out + 4 * FEAT, PB, PT);
  launch(cA1,  L1, L2, cA2, (unsigned long long)L2 * FEAT, FEAT, out + 3 * FEAT, PB, PT);
  launch(cA2,  L2, L3, cA3, (unsigned long long)L3 * FEAT, FEAT, out + 2 * FEAT, PB, PT);
  launch(cA3,  L3, L4, out + 0 * FEAT, PB, PT,                   out + 1 * FEAT, PB, PT);
}